// Middle_Integ_16097537425999
// MI455X (gfx1250) — compile-verified
//
#include <hip/hip_runtime.h>

#define SS 128
#define T_HIST 100
#define T_DATA 20000
#define NB 3

typedef float v2f __attribute__((ext_vector_type(2)));
typedef float v8f __attribute__((ext_vector_type(8)));
typedef int   v4i __attribute__((ext_vector_type(4)));

typedef __attribute__((address_space(1))) v4i* gas_v4i_ptr;   // global (AS1)
typedef __attribute__((address_space(3))) v4i* las_v4i_ptr;   // LDS (AS3)

__device__ __forceinline__ float sigmoid_fast(float x) {
    // sigmoid(x) = 1 / (1 + e^-x), via v_exp_f32 (exp2) + v_rcp_f32
    float e = __builtin_amdgcn_exp2f(-1.44269504088896f * x);
    return __builtin_amdgcn_rcpf(1.0f + e);
}

// Async global->LDS copy of one 16-byte chunk (CDNA5 GLOBAL_LOAD_ASYNC_TO_LDS_B128,
// tracked by ASYNCcnt). Builtin if available, else raw asm (portable per ISA doc).
__device__ __forceinline__ void async_copy_b128(const float* gsrc, float* ldst) {
#if __has_builtin(__builtin_amdgcn_global_load_async_to_lds_b128)
    __builtin_amdgcn_global_load_async_to_lds_b128(
        (gas_v4i_ptr)gsrc, (las_v4i_ptr)ldst, 0, 0);
#else
    unsigned lds_addr = (unsigned)(unsigned long long)ldst;  // low 32 bits = LDS byte offset
    asm volatile("global_load_async_to_lds_b128 %0, %1, off"
                 :: "v"(lds_addr), "v"(gsrc) : "memory");
#endif
}

__device__ __forceinline__ void wait_asynccnt0() {
#if __has_builtin(__builtin_amdgcn_s_wait_asynccnt)
    __builtin_amdgcn_s_wait_asynccnt(0);
#else
    asm volatile("s_wait_asynccnt 0" ::: "memory");
#endif
}

// ---------------------------------------------------------------------------
// Kernel 1: build the (unflipped, causal) temporal kernels, stored [j][s]
// k[s,j] = sum_b  tt*exp(-tt)*K[s,b],  tt = max(j-delta[s],0) * exp(-log_tau[b])
// ---------------------------------------------------------------------------
__global__ void build_kern_kernel(const float* __restrict__ K_spike,
                                  const float* __restrict__ tau_spike,
                                  const float* __restrict__ delta_spike,
                                  const float* __restrict__ tau_hist,
                                  const float* __restrict__ K_hist,
                                  const float* __restrict__ delta_hist,
                                  float* __restrict__ ancK,
                                  float* __restrict__ histK) {
    const int s = threadIdx.x;
    const float dsp = delta_spike[s];
    const float dhi = delta_hist[s];
    float invts[NB], invth[NB], ksp[NB], khi[NB];
    #pragma unroll
    for (int b = 0; b < NB; ++b) {
        invts[b] = __expf(-tau_spike[b]);
        invth[b] = __expf(-tau_hist[b]);
        ksp[b]   = K_spike[s * NB + b];
        khi[b]   = K_hist[s * NB + b];
    }
    for (int j = 0; j < T_HIST; ++j) {
        float ta = fmaxf((float)j - dsp, 0.0f);
        float th = fmaxf((float)j - dhi, 0.0f);
        float accA = 0.0f, accH = 0.0f;
        #pragma unroll
        for (int b = 0; b < NB; ++b) {
            float tta = ta * invts[b];
            accA = fmaf(tta * __expf(-tta), ksp[b], accA);
            float tth = th * invth[b];
            accH = fmaf(tth * __expf(-tth), khi[b], accH);
        }
        ancK[j * SS + s]  = accA;
        histK[j * SS + s] = accH;
    }
}

// ---------------------------------------------------------------------------
// Kernel 2: dual f32 WMMA GEMM (v_wmma_f32_16x16x4_f32, K=128 in 32 steps).
//   A_out[t,s]    = sum_c Z[t,c]  * C_den[s,c]
//   base_out[t,s] = sum_c Y[t,c]  * C_den[s,c] + S_conv[t,s] + theta_syn[s]
// ---------------------------------------------------------------------------
__global__ void gemm_wmma_kernel(const float* __restrict__ Z,
                                 const float* __restrict__ Y,
                                 const float* __restrict__ Sconv,
                                 const float* __restrict__ Cden,
                                 const float* __restrict__ theta_syn,
                                 float* __restrict__ A_out,
                                 float* __restrict__ base_out) {
    const int wave = threadIdx.x >> 5;      // 8 waves -> 8 s-tiles
    const int lane = threadIdx.x & 31;
    const int hi   = lane >> 4;             // 0 / 1
    const int r16  = lane & 15;
    const int t0   = blockIdx.x * 16;
    const int s0   = wave * 16;

    v8f accZ = {};
    v8f accY = {};

    const float* zrow = Z    + (size_t)(t0 + r16) * SS + 2 * hi;
    const float* yrow = Y    + (size_t)(t0 + r16) * SS + 2 * hi;
    const float* crow = Cden + (size_t)(s0 + r16) * SS + 2 * hi;

    #pragma unroll 8
    for (int c0 = 0; c0 < SS; c0 += 4) {
        v2f b  = *(const v2f*)(crow + c0);
        v2f az = *(const v2f*)(zrow + c0);
        v2f ay = *(const v2f*)(yrow + c0);
        accZ = __builtin_amdgcn_wmma_f32_16x16x4_f32(false, az, false, b,
                                                     (short)0, accZ, false, false);
        accY = __builtin_amdgcn_wmma_f32_16x16x4_f32(false, ay, false, b,
                                                     (short)0, accY, false, false);
    }

    const float th = theta_syn[s0 + r16];
    #pragma unroll
    for (int v = 0; v < 8; ++v) {
        int r = t0 + v + 8 * hi;        // C/D layout: M = v + 8*(lane>=16)
        int c = s0 + r16;               //             N = lane & 15
        size_t idx = (size_t)r * SS + c;
        A_out[idx]    = accZ[v];
        base_out[idx] = accY[v] + Sconv[idx] + th;
    }
}

// ---------------------------------------------------------------------------
// Kernel 3: causal depthwise FIR over time on A, accumulated into base.
//   base[t,s] += sum_{j=0..99} ancK[s,j] * A[t-1-j, s]   (A[<0] == 0)
// The 116-row halo window is flat-contiguous in both global memory and LDS,
// so staging is a pure async b128 flat copy (GLOBAL_LOAD_ASYNC_TO_LDS_B128 +
// s_wait_asynccnt), bypassing VGPRs. Compute reads are bank-conflict free
// (lane s -> bank s).
// ---------------------------------------------------------------------------
#define CTT 16
__global__ void depthwise_conv_kernel(const float* __restrict__ A,
                                      const float* __restrict__ ancK,
                                      float* __restrict__ base) {
    __shared__ float sA[(CTT + T_HIST) * SS];   // 116 * 128 floats = 59392 B
    const int s  = threadIdx.x;
    const int t0 = blockIdx.x * CTT;

    // zero-fill rows that correspond to t < 0 (first few blocks only)
    int r0 = 0;
    if (t0 < T_HIST) {
        r0 = T_HIST - t0;
        for (int r = 0; r < r0; ++r)
            sA[r * SS + s] = 0.0f;
    }

    // async flat copy of the valid region: rows [r0, 116) <- A[t0-100+r0 ...]
    {
        const float* src = A + (size_t)(t0 - T_HIST + r0) * SS;
        float*       dst = sA + r0 * SS;
        const int nflt = (CTT + T_HIST - r0) * SS;
        for (int i = threadIdx.x * 4; i < nflt; i += SS * 4)
            async_copy_b128(src + i, dst + i);
        wait_asynccnt0();
    }
    __syncthreads();

    float acc[CTT];
    #pragma unroll
    for (int i = 0; i < CTT; ++i) acc[i] = 0.0f;

    for (int j = 0; j < T_HIST; ++j) {
        float kj = ancK[j * SS + s];
        #pragma unroll
        for (int tt = 0; tt < CTT; ++tt)
            acc[tt] = fmaf(sA[(tt + (T_HIST - 1) - j) * SS + s], kj, acc[tt]);
    }

    #pragma unroll
    for (int tt = 0; tt < CTT; ++tt) {
        size_t idx = (size_t)(t0 + tt) * SS + s;
        base[idx] += acc[tt];
    }
}

// ---------------------------------------------------------------------------
// Kernel 4: the sequential recurrence. 128 independent per-channel scalar
// recurrences -> 4 blocks x 32 channels, 4 lanes/channel (25 taps each).
// z-history lives in a 128-slot LDS ring; stride 34 makes the (p, ch) lane
// access pattern bank-conflict free. All 4 parts of a channel sit in one
// wave, so the lag-1 store->load is ordered by per-wave in-order LDS; the
// 4-way partial sums reduce with intra-wave shuffles. No barriers in the loop.
// ---------------------------------------------------------------------------
#define PARTS 4
#define TAPS  25            // 100 / PARTS
#define RING  128
#define ZSTR  34
#define CHB   32            // channels per block

__global__ void scan_kernel(const float* __restrict__ base,
                            const float* __restrict__ noise,
                            const float* __restrict__ histK,
                            const float* __restrict__ W_sub,
                            const float* __restrict__ W_spike,
                            const float* __restrict__ theta_spike,
                            float* __restrict__ outY,
                            float* __restrict__ outZ,
                            float* __restrict__ outM0,
                            float* __restrict__ outM1) {
    __shared__ float zh[RING * ZSTR];           // 17408 B
    const int tid = threadIdx.x;                // 128
    const int chl = tid >> 2;                   // 0..31 local channel
    const int p   = tid & 3;                    // part
    const int ch  = blockIdx.x * CHB + chl;     // global channel

    for (int i = tid; i < RING * ZSTR; i += blockDim.x) zh[i] = 0.0f;

    float c[TAPS];
    #pragma unroll
    for (int i = 0; i < TAPS; ++i)
        c[i] = histK[(p * TAPS + i) * SS + ch];

    const float wsp = W_spike[ch];
    const float tsp = theta_spike[ch];
    const float wsb = W_sub[ch];
    const int jbase = p * TAPS;
    __syncthreads();

    for (int t = 0; t < T_DATA; ++t) {
        // fh_partial = sum_i zh[t-1-(jbase+i)] * c[i], 4 split accumulators
        float f0 = 0.f, f1 = 0.f, f2 = 0.f, f3 = 0.f;
        const int m0 = t - 1 - jbase;
        #pragma unroll
        for (int i = 0; i < TAPS; ++i) {
            int slot = (m0 - i) & (RING - 1);   // C '&' handles negatives
            float pr = zh[slot * ZSTR + chl] * c[i];
            if ((i & 3) == 0)      f0 += pr;
            else if ((i & 3) == 1) f1 += pr;
            else if ((i & 3) == 2) f2 += pr;
            else                   f3 += pr;
        }
        float fh = (f0 + f1) + (f2 + f3);
        fh += __shfl_xor(fh, 1, 32);
        fh += __shfl_xor(fh, 2, 32);

        const size_t idx = (size_t)t * SS + ch;
        __builtin_prefetch(base + idx + 16 * SS, 0, 0);   // global_prefetch

        float x    = sigmoid_fast(base[idx] + fh);
        float down = fmaf(x, wsp, tsp);
        float z    = sigmoid_fast(down + noise[idx]);     // sqrt(FIX_VAR)=1

        if (p == 3) zh[(t & (RING - 1)) * ZSTR + chl] = z;

        if (p == 0)      outY[idx]  = x * wsb;
        else if (p == 1) outZ[idx]  = z;
        else if (p == 2) outM0[idx] = down;
        else             outM1[idx] = down;
    }
}

// ---------------------------------------------------------------------------
extern "C" void kernel_launch(void* const* d_in, const int* in_sizes, int n_in,
                              void* d_out, int out_size, void* d_ws, size_t ws_size,
                              hipStream_t stream) {
    (void)in_sizes; (void)n_in; (void)out_size; (void)ws_size;

    const float* S_conv      = (const float*)d_in[0];
    const float* Y_ancest    = (const float*)d_in[1];
    const float* Z_ancest    = (const float*)d_in[2];
    const float* noise       = (const float*)d_in[3];
    const float* C_den       = (const float*)d_in[4];
    const float* W_sub       = (const float*)d_in[5];
    const float* theta_syn   = (const float*)d_in[6];
    const float* K_spike     = (const float*)d_in[7];
    const float* tau_spike   = (const float*)d_in[8];
    const float* delta_spike = (const float*)d_in[9];
    const float* theta_spike = (const float*)d_in[10];
    const float* W_spike     = (const float*)d_in[11];
    const float* tau_hist    = (const float*)d_in[12];
    const float* K_hist      = (const float*)d_in[13];
    const float* delta_hist  = (const float*)d_in[14];

    float* ws    = (float*)d_ws;
    float* A     = ws;                                    // T*S
    float* baseb = ws + (size_t)T_DATA * SS;              // T*S
    float* ancK  = ws + 2 * (size_t)T_DATA * SS;          // 100*S
    float* histK = ancK + (size_t)T_HIST * SS;            // 100*S

    float* out   = (float*)d_out;
    float* outY  = out;
    float* outZ  = out + 1 * (size_t)T_DATA * SS;
    float* outM0 = out + 2 * (size_t)T_DATA * SS;
    float* outM1 = out + 3 * (size_t)T_DATA * SS;

    build_kern_kernel<<<1, SS, 0, stream>>>(K_spike, tau_spike, delta_spike,
                                            tau_hist, K_hist, delta_hist,
                                            ancK, histK);
    gemm_wmma_kernel<<<T_DATA / 16, 256, 0, stream>>>(Z_ancest, Y_ancest, S_conv,
                                                      C_den, theta_syn, A, baseb);
    depthwise_conv_kernel<<<T_DATA / CTT, SS, 0, stream>>>(A, ancK, baseb);
    scan_kernel<<<SS / CHB, CHB * PARTS, 0, stream>>>(baseb, noise, histK,
                                                      W_sub, W_spike, theta_spike,
                                                      outY, outZ, outM0, outM1);
}